// HaarWaveletUnsqueeze_7387343749494
// MI455X (gfx1250) — compile-verified
//
#include <hip/hip_runtime.h>

// Haar wavelet unsqueeze (inverse 2x2 Haar synthesis) for MI455X / gfx1250.
//
// y   : (8, 256, 128, 128) f32 = (B=8, C=64, 4 planes[a,h,v,d], H=128, W=128)
// out : (8, 64, 256, 256)  f32
//   tl=.5(a+h+v+d) tr=.5(a-h+v-d) bl=.5(a+h-v-d) br=.5(a-h-v+d)
//   out[2i,2j]=tl out[2i,2j+1]=tr out[2i+1,2j]=bl out[2i+1,2j+1]=br
//
// HBM-bound (~0.5 FLOP/byte, 256 MiB total => ~11.5 us floor @ 23.3 TB/s).
// The 4x4 butterfly runs exactly (f32) on V_WMMA_F32_16X16X4_F32:
//   D(16x16) = A(16x4) x B(4x16); A[m,k] = 0.5*(-1)^popc((m&3)&k) (constants),
//   B[:,n] = (a,h,v,d) at position n. D's row replication means lanes 0-15
//   hold {tl,tr} and lanes 16-31 hold {bl,br} for the same 16 positions.
// Each wave processes one full input row (128 j = 4 tile-pairs): setup and
// addressing amortize, all loads/stores use one base + immediate offsets,
// 8 loads are issued up front for MLP. Streams are non-temporal (touched once).

typedef float v2f __attribute__((ext_vector_type(2)));
typedef float v4f __attribute__((ext_vector_type(4)));
typedef float v8f __attribute__((ext_vector_type(8)));

__global__ __launch_bounds__(256) void haar_unsqueeze_wmma(
    const float* __restrict__ y, float* __restrict__ out)
{
    constexpr int Wd = 128, Hd = 128, HW = Hd * Wd;

    const int tid  = blockIdx.x * blockDim.x + threadIdx.x;
    const int wave = tid >> 5;                 // one input row per wave
    const int lane = threadIdx.x & 31;
    const int l16  = lane & 15;                // N index within WMMA tile
    const int hi   = lane >> 4;                // 0: K=0,1 (a,h) | 1: K=2,3 (v,d)

    // wave -> (b, c, i)
    const int i = wave & 127;
    const int c = (wave >> 7) & 63;
    const int b = wave >> 13;

    // ---- A matrix: Haar synthesis coefficients (f32 16x4 layout:
    // lanes 0-15 VGPR0=K0,VGPR1=K1; lanes 16-31 VGPR0=K2,VGPR1=K3)
    const int r  = l16 & 3;
    const int k0 = hi << 1;
    const int k1 = k0 | 1;
    v2f A;
    A[0] = (__popc(r & k0) & 1) ? -0.5f : 0.5f;
    A[1] = (__popc(r & k1) & 1) ? -0.5f : 0.5f;

    // ---- issue all row loads up front: per lane a v2f (even/odd j pair)
    // from plane (2*hi) and plane (2*hi+1); constant offsets -> IOFFSET
    const float* p = y + ((size_t)((b * 256 + c * 4 + 2 * hi) * Hd + i) * Wd
                          + 2 * l16);
    v2f q0[4], q1[4];
#pragma unroll
    for (int t = 0; t < 4; ++t) {
        q0[t] = __builtin_nontemporal_load(
                    reinterpret_cast<const v2f*>(p + t * 32));        // a | v
        q1[t] = __builtin_nontemporal_load(
                    reinterpret_cast<const v2f*>(p + HW + t * 32));   // h | d
    }

    // ---- output base: lanes 0-15 -> row 2i, lanes 16-31 -> row 2i+1
    float* op = out + ((size_t)((b * 64 + c) * 256 + (2 * i + hi)) * 256
                       + 4 * l16);

    v8f cz = {};
#pragma unroll
    for (int t = 0; t < 4; ++t) {
        // B matrices (4x16 f32: VGPR0=K0|K2, VGPR1=K1|K3), two tiles per pair
        v2f B0; B0[0] = q0[t][0]; B0[1] = q1[t][0];   // even-j tile
        v2f B1; B1[0] = q0[t][1]; B1[1] = q1[t][1];   // odd-j tile

        v8f d0 = __builtin_amdgcn_wmma_f32_16x16x4_f32(
            false, A, false, B0, (short)0, cz, false, false);
        v8f d1 = __builtin_amdgcn_wmma_f32_16x16x4_f32(
            false, A, false, B1, (short)0, cz, false, false);

        // pack: hi=0 -> {tl0,tr0,tl1,tr1}, hi=1 -> {bl0,br0,bl1,br1}
        v4f o;
        o[0] = hi ? d0[2] : d0[0];
        o[1] = hi ? d0[3] : d0[1];
        o[2] = hi ? d1[2] : d1[0];
        o[3] = hi ? d1[3] : d1[1];

        __builtin_nontemporal_store(o, reinterpret_cast<v4f*>(op + t * 64));
    }
}

extern "C" void kernel_launch(void* const* d_in, const int* in_sizes, int n_in,
                              void* d_out, int out_size, void* d_ws, size_t ws_size,
                              hipStream_t stream) {
    (void)in_sizes; (void)n_in; (void)out_size; (void)d_ws; (void)ws_size;
    const float* y = (const float*)d_in[0];
    float* out     = (float*)d_out;

    // one wave per input row: 8 * 64 * 128 = 65536 waves
    const int total_waves = 8 * 64 * 128;
    const int threads     = 256;                           // 8 full waves/block
    const int blocks      = (total_waves * 32) / threads;  // 8192, exact cover
    haar_unsqueeze_wmma<<<blocks, threads, 0, stream>>>(y, out);
}